// FusedSparseLinear_70695161692194
// MI455X (gfx1250) — compile-verified
//
#include <hip/hip_runtime.h>
#include <hip/hip_bf16.h>

typedef float v2f __attribute__((ext_vector_type(2)));
typedef float v8f __attribute__((ext_vector_type(8)));
typedef int   v4i __attribute__((ext_vector_type(4)));

#define N_BLOCKS 1024
#define KDIM 4096
#define MDIM 4096
#define LDSS 68            // LDS row stride in floats: conflict-free & 16B aligned
#define TILE (64 * LDSS)   // floats per 64x64 tile buffer

#if defined(__has_builtin)
#if __has_builtin(__builtin_amdgcn_global_load_async_to_lds_b128)
#define HAVE_ASYNC_LDS 1
#endif
#endif
#ifndef HAVE_ASYNC_LDS
#define HAVE_ASYNC_LDS 0
#endif

// ---------------------------------------------------------------------------
// Pass 1: deterministic bucket sort of sparse blocks by row-block.
// ---------------------------------------------------------------------------
__global__ __launch_bounds__(N_BLOCKS) void bin_kernel(
    const int* __restrict__ block_rows,
    int* __restrict__ bucket_start,   // 65 ints
    int* __restrict__ sorted_idx)     // N_BLOCKS ints
{
    __shared__ int rows[N_BLOCKS];
    __shared__ int counts[64];
    __shared__ int starts[65];

    int tid = threadIdx.x;
    rows[tid] = block_rows[tid];
    if (tid < 64) counts[tid] = 0;
    __syncthreads();

    int r = rows[tid];
    atomicAdd(&counts[r], 1);
    __syncthreads();

    if (tid == 0) {
        int acc = 0;
        for (int i = 0; i < 64; ++i) { starts[i] = acc; acc += counts[i]; }
        starts[64] = acc;
    }
    __syncthreads();

    int rank = 0;
    for (int i = 0; i < tid; ++i) rank += (rows[i] == r) ? 1 : 0;
    sorted_idx[starts[r] + rank] = tid;

    if (tid < 65) bucket_start[tid] = starts[tid];
}

// ---------------------------------------------------------------------------
// Helpers
// ---------------------------------------------------------------------------
#if HAVE_ASYNC_LDS
__device__ __forceinline__ void wait_async_lds() {
#if __has_builtin(__builtin_amdgcn_s_wait_asynccnt)
    __builtin_amdgcn_s_wait_asynccnt(0);
#else
    asm volatile("s_wait_asynccnt 0" ::: "memory");
#endif
}

// one 16B async global->LDS transfer per lane (ASYNCcnt-tracked, no VGPR data)
__device__ __forceinline__ void async_cp16(const float* g, float* l) {
    __builtin_amdgcn_global_load_async_to_lds_b128(
        (__attribute__((address_space(1))) v4i*)g,
        (__attribute__((address_space(3))) v4i*)l,
        0, 0);
}

// issue async fills for one 64x64 x-tile + 64x64 values-tile
__device__ __forceinline__ void issue_tile_loads(
    const float* __restrict__ xsrc, const float* __restrict__ vsrc,
    float* xdst, float* vdst, int lrow, int lcol)
{
#pragma unroll
    for (int p = 0; p < 8; ++p) {
        const int rr = lrow + p * 8;
        async_cp16(xsrc + (size_t)rr * KDIM + lcol, &xdst[rr * LDSS + lcol]);
        async_cp16(vsrc + rr * 64 + lcol,           &vdst[rr * LDSS + lcol]);
    }
}
#endif

__device__ __forceinline__ void compute_tile(
    const float* __restrict__ xs, const float* __restrict__ vs,
    int w, int lm, int lh, v8f acc[4])
{
#pragma unroll 4
    for (int k0 = 0; k0 < 64; k0 += 4) {
        const int ka = k0 + 2 * lh;
        // A fragment: A[m][k], m = 16w+lm, K pair (ka, ka+1)
        v2f a;
        a.x = vs[(w * 16 + lm) * LDSS + ka];
        a.y = vs[(w * 16 + lm) * LDSS + ka + 1];
#pragma unroll
        for (int j = 0; j < 4; ++j) {
            // B fragment: B[k][n] = x[t0 + 16j + n][k], n = lm
            v2f b;
            b.x = xs[(j * 16 + lm) * LDSS + ka];
            b.y = xs[(j * 16 + lm) * LDSS + ka + 1];
            acc[j] = __builtin_amdgcn_wmma_f32_16x16x4_f32(
                false, a, false, b, (short)0, acc[j], false, false);
        }
    }
}

// ---------------------------------------------------------------------------
// Pass 2: double-buffered block-sparse GEMM, V_WMMA_F32_16X16X4_F32,
// fused bias + tanh-GELU epilogue. 4 waves; wave w owns M rows [16w,16w+16).
// ---------------------------------------------------------------------------
__global__ __launch_bounds__(128) void spmm_kernel(
    const float* __restrict__ x,            // [T, 4096]
    const float* __restrict__ values,       // [1024, 64, 64]
    const float* __restrict__ bias,         // [4096]
    const int*   __restrict__ block_cols,   // [1024]
    const int*   __restrict__ bucket_start, // [65]
    const int*   __restrict__ sorted_idx,   // [1024]
    float* __restrict__ out)                // [T, 4096]
{
    __shared__ float xsb[2][TILE];   // x tiles  [t][k]  (ping-pong)
    __shared__ float vsb[2][TILE];   // values   [m][k]  (ping-pong)

    const int t0   = blockIdx.x * 64;
    const int r    = blockIdx.y;
    const int tid  = threadIdx.x;
    const int lane = tid & 31;
    const int w    = tid >> 5;       // wave id 0..3
    const int lm   = lane & 15;
    const int lh   = lane >> 4;

    v8f acc[4] = {};

    const int beg = bucket_start[r];
    const int nb  = bucket_start[r + 1] - beg;

    const int lrow = tid >> 4;        // 0..7
    const int lcol = (tid & 15) * 4;  // 0..60

    // ---- prologue: fill buffer 0 with tile 0 ----
    if (nb > 0) {
        const int n = sorted_idx[beg];
        const int c = block_cols[n];
        const float* xsrc = x + (size_t)t0 * KDIM + (size_t)c * 64;
        const float* vsrc = values + (size_t)n * 4096;
#if HAVE_ASYNC_LDS
        issue_tile_loads(xsrc, vsrc, xsb[0], vsb[0], lrow, lcol);
        wait_async_lds();
#else
#pragma unroll
        for (int p = 0; p < 8; ++p) {
            const int rr = lrow + p * 8;
            *(float4*)&xsb[0][rr * LDSS + lcol] = *(const float4*)(xsrc + (size_t)rr * KDIM + lcol);
            *(float4*)&vsb[0][rr * LDSS + lcol] = *(const float4*)(vsrc + rr * 64 + lcol);
        }
#endif
    }
    __syncthreads();

    // ---- pipelined main loop: prefetch tile ib+1 while computing tile ib ----
    for (int ib = 0; ib < nb; ++ib) {
        const int cur = ib & 1;
        const int nxt = cur ^ 1;
        const bool have_next = (ib + 1) < nb;

        const float* xsrc = nullptr;
        const float* vsrc = nullptr;
        if (have_next) {
            const int n = sorted_idx[beg + ib + 1];
            const int c = block_cols[n];
            xsrc = x + (size_t)t0 * KDIM + (size_t)c * 64;
            vsrc = values + (size_t)n * 4096;
        }

#if HAVE_ASYNC_LDS
        if (have_next)
            issue_tile_loads(xsrc, vsrc, xsb[nxt], vsb[nxt], lrow, lcol);
        compute_tile(xsb[cur], vsb[cur], w, lm, lh, acc);
        wait_async_lds();
#else
        float4 xr[8], vr[8];
        if (have_next) {
#pragma unroll
            for (int p = 0; p < 8; ++p) {
                const int rr = lrow + p * 8;
                xr[p] = *(const float4*)(xsrc + (size_t)rr * KDIM + lcol);
                vr[p] = *(const float4*)(vsrc + rr * 64 + lcol);
            }
        }
        compute_tile(xsb[cur], vsb[cur], w, lm, lh, acc);
        if (have_next) {
#pragma unroll
            for (int p = 0; p < 8; ++p) {
                const int rr = lrow + p * 8;
                *(float4*)&xsb[nxt][rr * LDSS + lcol] = xr[p];
                *(float4*)&vsb[nxt][rr * LDSS + lcol] = vr[p];
            }
        }
#endif
        __syncthreads();
    }

    // ---- epilogue: stage D tiles through LDS for coalesced bias+GELU store ----
    float* ytile = xsb[0];
#pragma unroll
    for (int j = 0; j < 4; ++j) {
#pragma unroll
        for (int v = 0; v < 8; ++v) {
            const int ml = v + 8 * lh;        // local M row 0..15
            const int tt = j * 16 + lm;       // local t  0..63
            ytile[tt * LDSS + w * 16 + ml] = acc[j][v];
        }
    }
    __syncthreads();

    const float* bptr = bias + r * 64;
#pragma unroll 4
    for (int p = 0; p < 32; ++p) {
        const int idx = p * 128 + tid;
        const int tt = idx >> 6;
        const int mm = idx & 63;
        float vvv = ytile[tt * LDSS + mm] + bptr[mm];
        float inner = 0.7978845608f * (vvv + 0.044715f * vvv * vvv * vvv);
        float yv = 0.5f * vvv * (1.0f + tanhf(inner));
        out[(size_t)(t0 + tt) * MDIM + r * 64 + mm] = yv;
    }
}

extern "C" void kernel_launch(void* const* d_in, const int* in_sizes, int n_in,
                              void* d_out, int out_size, void* d_ws, size_t ws_size,
                              hipStream_t stream) {
    const float* x          = (const float*)d_in[0];
    const float* values     = (const float*)d_in[1];
    const float* bias       = (const float*)d_in[2];
    const int*   block_rows = (const int*)d_in[3];
    const int*   block_cols = (const int*)d_in[4];
    float* out = (float*)d_out;

    const int T = in_sizes[0] / KDIM;   // 2048

    int* bucket_start = (int*)d_ws;          // 65 ints
    int* sorted_idx   = bucket_start + 65;   // 1024 ints

    bin_kernel<<<1, N_BLOCKS, 0, stream>>>(block_rows, bucket_start, sorted_idx);

    dim3 grid(T / 64, MDIM / 64);            // (32, 64)
    spmm_kernel<<<grid, 128, 0, stream>>>(x, values, bias, block_cols,
                                          bucket_start, sorted_idx, out);
}